// SPFNSAttention_27814208209734
// MI455X (gfx1250) — compile-verified
//
#include <hip/hip_runtime.h>
#include <hip/hip_bf16.h>
#include <math.h>

// Problem constants (from reference)
#define BBATCH 2
#define SSEQ   2048
#define EEMB   768
#define HHEADS 12
#define DHEAD  64
#define MROWS  (BBATCH*SSEQ)    // 4096
#define BH     (BBATCH*HHEADS)  // 24

typedef __attribute__((ext_vector_type(2))) float v2f;
typedef __attribute__((ext_vector_type(8))) float v8f;
typedef __attribute__((ext_vector_type(4))) unsigned u32x4;
typedef __attribute__((ext_vector_type(8))) int i32x8;
typedef __attribute__((ext_vector_type(4))) int i32x4;

// CDNA5 fp32 WMMA: D(16x16) = A(16x4) x B(4x16) + C
static __device__ __forceinline__ v8f wmma4(v2f a, v2f b, v8f c) {
    return __builtin_amdgcn_wmma_f32_16x16x4_f32(false, a, false, b, (short)0, c, false, false);
}

// ---------------------------------------------------------------------------
// Tensor Data Mover: stage a (tile0 x tile1) fp32 tile (row stride `stride0`
// dwords in memory) into LDS, inserting a 2-DWORD pad after every row of
// tile0 dwords (pad_interval code: tile0=32 -> 4, tile0=64 -> 5).
// tile1==0 => 1-D copy of tile0 dwords (pads still applied per interval).
// Issued by one wave; completion via s_wait_tensorcnt + workgroup barrier.
// ---------------------------------------------------------------------------
static __device__ __forceinline__ void tdm_load_tile(
    unsigned lds_off, const void* gaddr, unsigned tile0, unsigned tile1,
    unsigned stride0, unsigned pad_iv)
{
    unsigned long long ga = (unsigned long long)(size_t)gaddr;
    u32x4 g0;
    g0.x = 1u;                                            // count=1 (valid), user mode
    g0.y = lds_off;                                       // lds_addr (bytes)
    g0.z = (unsigned)(ga & 0xFFFFFFFFu);                  // global_addr[31:0]
    g0.w = (unsigned)((ga >> 32) & 0x01FFFFFFu) | (2u << 30); // addr[56:32] | type=2
    const unsigned td0 = 0x7FFFFFFFu;                     // huge tensor dims: no OOB clip
    i32x8 g1;
    g1[0] = (int)((2u << 16) | (1u << 20) | (pad_iv << 22) | (1u << 25));
    //         data_size=4B   pad_enable    pad_interval      pad_amount=2 DWORDs
    g1[1] = (int)((td0 & 0xFFFFu) << 16);                 // tensor_dim0[15:0]
    g1[2] = (int)((td0 >> 16) | (1u << 16));              // tensor_dim0[31:16] | tensor_dim1[15:0]=1
    g1[3] = (int)(tile0 << 16);                           // tensor_dim1[31:16]=0 | tile_dim0
    g1[4] = (int)(tile1 & 0xFFFFu);                       // tile_dim1 (0 => 1-D) | tile_dim2=0
    g1[5] = (int)stride0;                                 // tensor_dim0_stride[31:0]
    g1[6] = 0;                                            // stride0 hi | dim1_stride lo
    g1[7] = 0;
    i32x4 z4 = {};
#if defined(__clang_major__) && __clang_major__ >= 23
    i32x8 z8 = {};
    __builtin_amdgcn_tensor_load_to_lds(g0, g1, z4, z4, z8, 0);
#else
    __builtin_amdgcn_tensor_load_to_lds(g0, g1, z4, z4, 0);
#endif
}

// ---------------------------------------------------------------------------
// 1) Row L2 norm over 768 columns (hidden_states -> hs)
// ---------------------------------------------------------------------------
__global__ __launch_bounds__(256) void l2norm_rows_kernel(
    const float* __restrict__ x, float* __restrict__ y, int ncols)
{
    const int row = blockIdx.x, tid = threadIdx.x;
    const float* xr = x + (size_t)row * ncols;
    float s = 0.f;
    for (int i = tid; i < ncols; i += 256) { float v = xr[i]; s += v * v; }
    #pragma unroll
    for (int m = 16; m >= 1; m >>= 1) s += __shfl_xor(s, m, 32);
    __shared__ float red[9];
    if ((tid & 31) == 0) red[tid >> 5] = s;
    __syncthreads();
    if (tid == 0) {
        float t = 0.f;
        for (int i = 0; i < 8; i++) t += red[i];
        red[8] = 1.0f / fmaxf(sqrtf(t), 1e-12f);
    }
    __syncthreads();
    const float inv = red[8];
    for (int i = tid; i < ncols; i += 256) y[(size_t)row * ncols + i] = xr[i] * inv;
}

// ---------------------------------------------------------------------------
// 2) WMMA GEMM: out = A[4096x768] @ W[768x768] + bias (+ resid).
//    64x128 block tile, 8 waves (4M x 2N), wave = 16x64 via 4 accumulators.
//    A tile staged by TDM (2-D, 32x64, pad to 34-dword rows); W tile staged
//    cooperatively (needs transpose). head_major=1 writes [B,H,S,D].
// ---------------------------------------------------------------------------
__global__ __launch_bounds__(256) void gemm768_kernel(
    const float* __restrict__ A, const float* __restrict__ W,
    const float* __restrict__ bias, const float* __restrict__ resid,
    float* __restrict__ out, int head_major)
{
    __shared__ float Xt[64 * 34];    // A tile 64x32, rows padded to 34 (TDM pad)
    __shared__ float WT[128 * 34];   // W tile transposed: WT[n][k]
    const int tid  = threadIdx.x;
    const int wave = tid >> 5, lane = tid & 31;
    const int wm = wave & 3, wn = wave >> 2;
    const int half = lane >> 4, l16 = lane & 15;
    const int h2 = half * 2;
    const int mb = blockIdx.x * 64, nb = blockIdx.y * 128;
    const unsigned xt_lds = (unsigned)(size_t)&Xt[0];

    v8f acc[4] = {};
    for (int k0 = 0; k0 < 768; k0 += 32) {
        if (k0) __syncthreads();                       // prior consumers done
        if (wave == 0)
            tdm_load_tile(xt_lds, A + (size_t)mb * 768 + k0, 32, 64, 768, 4);
        #pragma unroll
        for (int i = 0; i < 16; i++) {                 // 32x128 W tile -> transposed
            int e = tid + i * 256, kk = e >> 7, n = e & 127;
            WT[n * 34 + kk] = W[(size_t)(k0 + kk) * 768 + nb + n];
        }
        if (wave == 0) __builtin_amdgcn_s_wait_tensorcnt(0);
        __syncthreads();
        const int arow = (wm * 16 + l16) * 34;
        #pragma unroll
        for (int kk = 0; kk < 32; kk += 8) {           // preload frags, then WMMA
            v2f a0 = *(const v2f*)&Xt[arow + kk + h2];
            v2f a1 = *(const v2f*)&Xt[arow + kk + 4 + h2];
            v2f b0[4], b1[4];
            #pragma unroll
            for (int t = 0; t < 4; t++) {
                const int brow = (wn * 64 + t * 16 + l16) * 34;
                b0[t] = *(const v2f*)&WT[brow + kk + h2];
                b1[t] = *(const v2f*)&WT[brow + kk + 4 + h2];
            }
            #pragma unroll
            for (int t = 0; t < 4; t++) acc[t] = wmma4(a0, b0[t], acc[t]);
            #pragma unroll
            for (int t = 0; t < 4; t++) acc[t] = wmma4(a1, b1[t], acc[t]);
        }
    }
    #pragma unroll
    for (int t = 0; t < 4; t++) {
        const int ncol = nb + wn * 64 + t * 16 + l16;
        const float bval = bias[ncol];
        #pragma unroll
        for (int r = 0; r < 8; r++) {
            const int row = mb + wm * 16 + half * 8 + r;
            float val = acc[t][r] + bval;
            if (resid) val += resid[(size_t)row * 768 + ncol];
            if (head_major) {
                const int h = ncol >> 6, d = ncol & 63;
                const int bb = row >> 11, s = row & (SSEQ - 1);
                out[(((size_t)(bb * HHEADS + h)) * SSEQ + s) * DHEAD + d] = val;
            } else {
                out[(size_t)row * 768 + ncol] = val;
            }
        }
    }
}

// ---------------------------------------------------------------------------
// 3) Per-head L2 norm: rows of 64 (one wave per row, 8 rows per block)
// ---------------------------------------------------------------------------
__global__ __launch_bounds__(256) void headnorm_kernel(float* __restrict__ x)
{
    const int wave = threadIdx.x >> 5, lane = threadIdx.x & 31;
    const int row = blockIdx.x * 8 + wave;
    float* r = x + (size_t)row * 64;
    float a = r[lane], b = r[lane + 32];
    float s = a * a + b * b;
    #pragma unroll
    for (int m = 16; m >= 1; m >>= 1) s += __shfl_xor(s, m, 32);
    const float inv = 1.0f / fmaxf(sqrtf(s), 1e-12f);
    r[lane] = a * inv;
    r[lane + 32] = b * inv;
}

// ---------------------------------------------------------------------------
// 4) Pass A: column sums of score = exp(-acos(clamp(q.k))^2). WMMA QK^T.
//    Q/K tiles staged by TDM (1-D contiguous, hardware row padding).
//    grid: (S/64 query tiles, S/128 key tiles, B*H)
// ---------------------------------------------------------------------------
__global__ __launch_bounds__(256) void colsum_kernel(
    const float* __restrict__ q, const float* __restrict__ k,
    float* __restrict__ colsum)
{
    __shared__ float Qt[64 * 66];
    __shared__ float Kt[128 * 66];
    const int tid  = threadIdx.x;
    const int wave = tid >> 5, lane = tid & 31;
    const int wm = wave & 3, wn = wave >> 2;
    const int half = lane >> 4, l16 = lane & 15;
    const int h2 = half * 2;
    const int mb = blockIdx.x * 64, nb = blockIdx.y * 128, bh = blockIdx.z;
    const float* Qh = q + (size_t)bh * SSEQ * DHEAD;
    const float* Kh = k + (size_t)bh * SSEQ * DHEAD;

    if (wave == 0) {
        tdm_load_tile((unsigned)(size_t)&Qt[0], Qh + (size_t)mb * 64, 64 * 64, 0, 64, 5);
        tdm_load_tile((unsigned)(size_t)&Kt[0], Kh + (size_t)nb * 64, 128 * 64, 0, 64, 5);
        __builtin_amdgcn_s_wait_tensorcnt(0);
    }
    __syncthreads();

    v8f acc[4] = {};
    const int arow = (wm * 16 + l16) * 66;
    #pragma unroll
    for (int kk = 0; kk < 64; kk += 8) {
        v2f a0 = *(const v2f*)&Qt[arow + kk + h2];
        v2f a1 = *(const v2f*)&Qt[arow + kk + 4 + h2];
        v2f b0[4], b1[4];
        #pragma unroll
        for (int t = 0; t < 4; t++) {
            const int brow = (wn * 64 + t * 16 + l16) * 66;
            b0[t] = *(const v2f*)&Kt[brow + kk + h2];
            b1[t] = *(const v2f*)&Kt[brow + kk + 4 + h2];
        }
        #pragma unroll
        for (int t = 0; t < 4; t++) acc[t] = wmma4(a0, b0[t], acc[t]);
        #pragma unroll
        for (int t = 0; t < 4; t++) acc[t] = wmma4(a1, b1[t], acc[t]);
    }
    #pragma unroll
    for (int t = 0; t < 4; t++) {
        float s = 0.f;
        #pragma unroll
        for (int r = 0; r < 8; r++) {
            float c = fminf(fmaxf(acc[t][r], -1.0f + 1e-7f), 1.0f - 1e-7f);
            float g = acosf(c);
            s += __expf(-g * g);
        }
        s += __shfl_xor(s, 16, 32);   // combine row-halves (same column)
        if (half == 0)
            atomicAdd(&colsum[(size_t)bh * SSEQ + nb + wn * 64 + t * 16 + l16], s);
    }
}

// 5) colsum -> colsum^(-1/2)
__global__ void colscale_kernel(float* __restrict__ cs, int n)
{
    int i = blockIdx.x * blockDim.x + threadIdx.x;
    if (i < n) cs[i] = rsqrtf(fmaxf(cs[i], 1e-30f));
}

// ---------------------------------------------------------------------------
// 6) Pass B (fused): recompute scores tile-by-tile, scale by col^-1/2,
//    accumulate row denominators and O += P@V via WMMA. row^-a cancels in
//    the L1 normalization so it is dropped. Q/K tiles staged by TDM; V tile
//    staged cooperatively (needs transpose).
//    grid: (S/64 query tiles, B*H). Block owns 64 queries x full head.
// ---------------------------------------------------------------------------
__global__ __launch_bounds__(256) void attn_av_kernel(
    const float* __restrict__ q, const float* __restrict__ k,
    const float* __restrict__ v, const float* __restrict__ cs,
    float* __restrict__ attn_out)
{
    __shared__ float Qt[64 * 66];
    __shared__ float Kt[128 * 66];
    __shared__ float VT[64 * 130];   // VT[d][t] (transposed V tile)
    __shared__ float Pt[64 * 130];   // activated probabilities
    __shared__ float rsum[64];
    const int tid  = threadIdx.x;
    const int wave = tid >> 5, lane = tid & 31;
    const int wm = wave & 3, wn = wave >> 2;
    const int half = lane >> 4, l16 = lane & 15;
    const int h2 = half * 2;
    const int mb = blockIdx.x * 64, bh = blockIdx.y;
    const int bidx = bh / HHEADS, h = bh % HHEADS;
    const float* Qh = q + (size_t)bh * SSEQ * DHEAD;
    const float* Kh = k + (size_t)bh * SSEQ * DHEAD;
    const float* Vh = v + (size_t)bh * SSEQ * DHEAD;
    const unsigned kt_lds = (unsigned)(size_t)&Kt[0];

    if (wave == 0)
        tdm_load_tile((unsigned)(size_t)&Qt[0], Qh + (size_t)mb * 64, 64 * 64, 0, 64, 5);

    v8f oacc[2] = {};
    float rden[8] = {};

    for (int nb = 0; nb < SSEQ; nb += 128) {
        if (nb) __syncthreads();     // prior iteration's consumers done
        if (wave == 0)
            tdm_load_tile(kt_lds, Kh + (size_t)nb * 64, 128 * 64, 0, 64, 5);
        #pragma unroll
        for (int i = 0; i < 32; i++) {
            int e = tid + i * 256, tt = e >> 6, c = e & 63;
            VT[c * 130 + tt] = Vh[(size_t)(nb + tt) * 64 + c];
        }
        if (wave == 0) __builtin_amdgcn_s_wait_tensorcnt(0);  // covers Qt too
        __syncthreads();

        // P = Q @ K^T (64x128 per block; 16x64 per wave)
        v8f pacc[4] = {};
        const int arow = (wm * 16 + l16) * 66;
        #pragma unroll
        for (int kk = 0; kk < 64; kk += 8) {
            v2f a0 = *(const v2f*)&Qt[arow + kk + h2];
            v2f a1 = *(const v2f*)&Qt[arow + kk + 4 + h2];
            v2f b0[4], b1[4];
            #pragma unroll
            for (int t = 0; t < 4; t++) {
                const int brow = (wn * 64 + t * 16 + l16) * 66;
                b0[t] = *(const v2f*)&Kt[brow + kk + h2];
                b1[t] = *(const v2f*)&Kt[brow + kk + 4 + h2];
            }
            #pragma unroll
            for (int t = 0; t < 4; t++) pacc[t] = wmma4(a0, b0[t], pacc[t]);
            #pragma unroll
            for (int t = 0; t < 4; t++) pacc[t] = wmma4(a1, b1[t], pacc[t]);
        }
        // activation + column scaling + row partial sums; stash P in LDS
        #pragma unroll
        for (int t = 0; t < 4; t++) {
            const int col = nb + wn * 64 + t * 16 + l16;
            const float csv = cs[(size_t)bh * SSEQ + col];
            #pragma unroll
            for (int r = 0; r < 8; r++) {
                float c = fminf(fmaxf(pacc[t][r], -1.0f + 1e-7f), 1.0f - 1e-7f);
                float g = acosf(c);
                float p = __expf(-g * g) * csv;
                rden[r] += p;
                Pt[(wm * 16 + half * 8 + r) * 130 + wn * 64 + t * 16 + l16] = p;
            }
        }
        __syncthreads();

        // O += P[64x128] @ V[128x64]; wave covers 16 rows x 32 cols (2 accum)
        const int prow = (wm * 16 + l16) * 130;
        #pragma unroll
        for (int kk = 0; kk < 128; kk += 8) {
            v2f a0 = *(const v2f*)&Pt[prow + kk + h2];
            v2f a1 = *(const v2f*)&Pt[prow + kk + 4 + h2];
            v2f b0[2], b1[2];
            #pragma unroll
            for (int t = 0; t < 2; t++) {
                const int brow = (wn * 32 + t * 16 + l16) * 130;
                b0[t] = *(const v2f*)&VT[brow + kk + h2];
                b1[t] = *(const v2f*)&VT[brow + kk + 4 + h2];
            }
            #pragma unroll
            for (int t = 0; t < 2; t++) oacc[t] = wmma4(a0, b0[t], oacc[t]);
            #pragma unroll
            for (int t = 0; t < 2; t++) oacc[t] = wmma4(a1, b1[t], oacc[t]);
        }
    }

    // full row denominators: reduce over 16 lanes per half, then across wn
    #pragma unroll
    for (int r = 0; r < 8; r++) {
        float s = rden[r];
        s += __shfl_xor(s, 1, 32); s += __shfl_xor(s, 2, 32);
        s += __shfl_xor(s, 4, 32); s += __shfl_xor(s, 8, 32);
        rden[r] = s;
    }
    if (wn == 0 && l16 == 0) {
        #pragma unroll
        for (int r = 0; r < 8; r++) rsum[wm * 16 + half * 8 + r] = rden[r];
    }
    __syncthreads();
    if (wn == 1 && l16 == 0) {
        #pragma unroll
        for (int r = 0; r < 8; r++) rsum[wm * 16 + half * 8 + r] += rden[r];
    }
    __syncthreads();

    #pragma unroll
    for (int t = 0; t < 2; t++) {
        const int d = wn * 32 + t * 16 + l16;
        #pragma unroll
        for (int r = 0; r < 8; r++) {
            const int rloc = wm * 16 + half * 8 + r;
            const float inv = 1.0f / fmaxf(rsum[rloc], 1e-12f);
            const int s = mb + rloc;
            attn_out[((size_t)(bidx * SSEQ + s)) * 768 + h * 64 + d] = oacc[t][r] * inv;
        }
    }
}

// ---------------------------------------------------------------------------
// 7) LayerNorm over rows of 768
// ---------------------------------------------------------------------------
__global__ __launch_bounds__(256) void layernorm_kernel(
    const float* __restrict__ y, const float* __restrict__ g,
    const float* __restrict__ bta, float* __restrict__ out)
{
    const int row = blockIdx.x, tid = threadIdx.x;
    const float* yr = y + (size_t)row * 768;
    float vv[3], s = 0.f, s2 = 0.f;
    #pragma unroll
    for (int i = 0; i < 3; i++) {
        vv[i] = yr[tid + i * 256];
        s += vv[i]; s2 += vv[i] * vv[i];
    }
    #pragma unroll
    for (int m = 16; m >= 1; m >>= 1) { s += __shfl_xor(s, m, 32); s2 += __shfl_xor(s2, m, 32); }
    __shared__ float rs[8], rs2[8], bc[2];
    if ((tid & 31) == 0) { rs[tid >> 5] = s; rs2[tid >> 5] = s2; }
    __syncthreads();
    if (tid == 0) {
        float ts = 0.f, ts2 = 0.f;
        for (int i = 0; i < 8; i++) { ts += rs[i]; ts2 += rs2[i]; }
        const float mu = ts / 768.0f;
        const float var = ts2 / 768.0f - mu * mu;
        bc[0] = mu; bc[1] = rsqrtf(var + 1e-12f);
    }
    __syncthreads();
    const float mu = bc[0], inv = bc[1];
    #pragma unroll
    for (int i = 0; i < 3; i++) {
        const int c = tid + i * 256;
        out[(size_t)row * 768 + c] = (vv[i] - mu) * inv * g[c] + bta[c];
    }
}

// ---------------------------------------------------------------------------
extern "C" void kernel_launch(void* const* d_in, const int* in_sizes, int n_in,
                              void* d_out, int out_size, void* d_ws, size_t ws_size,
                              hipStream_t stream)
{
    (void)in_sizes; (void)n_in; (void)out_size; (void)ws_size;
    const float* hid = (const float*)d_in[0];
    const float* Wq  = (const float*)d_in[1];
    const float* bq  = (const float*)d_in[2];
    const float* Wk  = (const float*)d_in[3];
    const float* bk  = (const float*)d_in[4];
    const float* Wv  = (const float*)d_in[5];
    const float* bv  = (const float*)d_in[6];
    const float* Wo  = (const float*)d_in[7];
    const float* bo  = (const float*)d_in[8];
    const float* lg  = (const float*)d_in[9];
    const float* lb  = (const float*)d_in[10];
    float* out = (float*)d_out;

    float* ws = (float*)d_ws;
    const size_t NME = (size_t)MROWS * EEMB;   // 3,145,728 floats
    float* hs   = ws;
    float* qb   = hs + NME;
    float* kb   = qb + NME;
    float* vb   = kb + NME;
    float* attn = vb + NME;
    float* yb   = attn + NME;
    float* csum = yb + NME;                    // BH*SSEQ = 49152 floats

    l2norm_rows_kernel<<<MROWS, 256, 0, stream>>>(hid, hs, EEMB);

    dim3 gg(MROWS / 64, EEMB / 128);
    gemm768_kernel<<<gg, 256, 0, stream>>>(hs, Wq, bq, nullptr, qb, 1);
    gemm768_kernel<<<gg, 256, 0, stream>>>(hs, Wk, bk, nullptr, kb, 1);
    gemm768_kernel<<<gg, 256, 0, stream>>>(hs, Wv, bv, nullptr, vb, 1);

    headnorm_kernel<<<(MROWS * HHEADS) / 8, 256, 0, stream>>>(qb);
    headnorm_kernel<<<(MROWS * HHEADS) / 8, 256, 0, stream>>>(kb);

    hipMemsetAsync(csum, 0, (size_t)BH * SSEQ * sizeof(float), stream);
    colsum_kernel<<<dim3(SSEQ / 64, SSEQ / 128, BH), 256, 0, stream>>>(qb, kb, csum);
    colscale_kernel<<<(BH * SSEQ + 255) / 256, 256, 0, stream>>>(csum, BH * SSEQ);

    attn_av_kernel<<<dim3(SSEQ / 64, BH), 256, 0, stream>>>(qb, kb, vb, csum, attn);

    gemm768_kernel<<<gg, 256, 0, stream>>>(attn, Wo, bo, hid, yb, 0);
    layernorm_kernel<<<MROWS, 256, 0, stream>>>(yb, lg, lb, out);
}